// VectorQuantizerProjection_36429912605193
// MI455X (gfx1250) — compile-verified
//
#include <hip/hip_runtime.h>
#include <hip/hip_bf16.h>

// Problem constants (fixed by the reference)
#define NTOK   16384      // 16*1024 tokens
#define EDIM   256        // embedding dim
#define PDIM   128        // projection dim
#define NCODE  8192       // codebook entries

typedef __attribute__((ext_vector_type(2))) float v2f;
typedef __attribute__((ext_vector_type(8))) float v8f;

// ---------------------------------------------------------------------------
// Kernel 1: project both token rows and codebook rows into P-space.
// rows [0, NTOK) -> xp, rows [NTOK, NTOK+NCODE) -> cbp.
// One block per row (128 threads = one output column each); the row data is
// block-uniform so the compiler turns those loads into scalar loads.
// ---------------------------------------------------------------------------
__global__ __launch_bounds__(PDIM) void vq_project_kernel(
    const float* __restrict__ x, const float* __restrict__ cb,
    const float* __restrict__ w, const float* __restrict__ bias,
    float* __restrict__ xp, float* __restrict__ cbp) {
  const int row = blockIdx.x;
  const int p = threadIdx.x;
  const float* src = (row < NTOK) ? (x + (size_t)row * EDIM)
                                  : (cb + (size_t)(row - NTOK) * EDIM);
  float acc = bias[p];
#pragma unroll 8
  for (int d = 0; d < EDIM; ++d) acc = fmaf(src[d], w[d * PDIM + p], acc);
  if (row < NTOK) xp[(size_t)row * PDIM + p] = acc;
  else            cbp[(size_t)(row - NTOK) * PDIM + p] = acc;
}

// ---------------------------------------------------------------------------
// Kernel 2: squared norms of projected codebook rows. One wave32 per row.
// ---------------------------------------------------------------------------
__global__ __launch_bounds__(32) void vq_cc_kernel(const float* __restrict__ cbp,
                                                   float* __restrict__ cc) {
  const int row = blockIdx.x;
  const int l = threadIdx.x;
  float s = 0.f;
#pragma unroll
  for (int i = l; i < PDIM; i += 32) {
    float v = cbp[(size_t)row * PDIM + i];
    s = fmaf(v, v, s);
  }
#pragma unroll
  for (int m = 16; m; m >>= 1) s += __shfl_xor(s, m, 32);
  if (l == 0) cc[row] = s;
}

// ---------------------------------------------------------------------------
// Kernel 3: fused distance GEMM + argmin via V_WMMA_F32_16X16X4_F32.
// Block = 256 threads (8 waves). Each wave owns 16 token rows; its 16x128
// A-panel is preloaded as 32 f32-WMMA A-fragments (64 VGPRs). Codebook
// columns stream through LDS in 64-col x 128-K chunks shared by all waves.
// argmin(d) == argmin(cc_k - 2 * <x,c_k>) so the x-norm term is dropped.
// ---------------------------------------------------------------------------
#define CHUNK_COLS 64
#define LDS_STRIDE 132   // 132 % 64 == 4 -> 16 cols hit 16 distinct banks; 16B aligned

__global__ __launch_bounds__(256) void vq_argmin_kernel(
    const float* __restrict__ xp, const float* __restrict__ cbp,
    const float* __restrict__ cc, int* __restrict__ idx) {
  __shared__ float lds_b[CHUNK_COLS * LDS_STRIDE];
  __shared__ float lds_cc[CHUNK_COLS];

  const int t = threadIdx.x;
  const int lane = t & 31;
  const int wave = t >> 5;
  const int m = lane & 15;               // row within 16-row block / col within tile
  const int khalf = (lane >> 4) << 1;    // 0 for lanes 0-15, 2 for lanes 16-31
  const int rowBase = blockIdx.x * 128 + wave * 16;

  // Preload A fragments: afr[s] covers K = 4*s .. 4*s+3 of this wave's 16 rows.
  v2f afr[32];
  {
    const float* arow = xp + (size_t)(rowBase + m) * PDIM + khalf;
#pragma unroll
    for (int s = 0; s < 32; ++s) afr[s] = *(const v2f*)(arow + s * 4);
  }

  float minv[8];
  int mini[8];
#pragma unroll
  for (int r = 0; r < 8; ++r) { minv[r] = 3.4e38f; mini[r] = 0; }

  for (int chunk = 0; chunk < NCODE / CHUNK_COLS; ++chunk) {
    const int colBase = chunk * CHUNK_COLS;
    // Cooperative load: 64 cols x 128 K floats. 4 threads per column,
    // 8 float4 each. LDS layout [col][k] with padded stride.
    {
      const int tcol = t >> 2, part = t & 3;
      const float4* src = (const float4*)(cbp + (size_t)(colBase + tcol) * PDIM);
      float4* dst = (float4*)(&lds_b[tcol * LDS_STRIDE]);
#pragma unroll
      for (int j = 0; j < 8; ++j) dst[part * 8 + j] = src[part * 8 + j];
      if (t < CHUNK_COLS) lds_cc[t] = cc[colBase + t];
    }
    __syncthreads();

#pragma unroll
    for (int tile = 0; tile < CHUNK_COLS / 16; ++tile) {
      const float* bcol = &lds_b[(tile * 16 + m) * LDS_STRIDE + khalf];
      v8f c = {};
#pragma unroll
      for (int s = 0; s < 32; ++s) {
        v2f b = *(const v2f*)(bcol + s * 4);
        c = __builtin_amdgcn_wmma_f32_16x16x4_f32(
            /*neg_a=*/false, afr[s], /*neg_b=*/false, b,
            /*c_mod=*/(short)0, c, /*reuse_a=*/false, /*reuse_b=*/false);
      }
      const int col = colBase + tile * 16 + m;
      const float ccv = lds_cc[tile * 16 + m];
#pragma unroll
      for (int r = 0; r < 8; ++r) {
        float v = fmaf(-2.f, c[r], ccv);   // cc_k - 2 * dot
        if (v < minv[r]) { minv[r] = v; mini[r] = col; }
      }
    }
    __syncthreads();
  }

  // Reduce across the 16 lanes of each half-wave (independent rows per half).
#pragma unroll
  for (int r = 0; r < 8; ++r) {
    float v = minv[r];
    int i = mini[r];
#pragma unroll
    for (int mm = 8; mm; mm >>= 1) {
      float ov = __shfl_xor(v, mm, 16);
      int oi = __shfl_xor(i, mm, 16);
      if (ov < v || (ov == v && oi < i)) { v = ov; i = oi; }
    }
    if (m == 0) idx[rowBase + r + ((lane >> 4) << 3)] = i;
  }
}

// ---------------------------------------------------------------------------
// Kernel 4: zero-fill the one-hot region (537 MB; the HBM-store roofline).
// ---------------------------------------------------------------------------
__global__ __launch_bounds__(256) void vq_fill_kernel(float4* __restrict__ out,
                                                      long long n4) {
  const long long stride = (long long)gridDim.x * blockDim.x;
  const float4 z = make_float4(0.f, 0.f, 0.f, 0.f);
  for (long long i = (long long)blockIdx.x * blockDim.x + threadIdx.x; i < n4;
       i += stride)
    out[i] = z;
}

// ---------------------------------------------------------------------------
// Kernel 5: scatter the ones + gather quantized rows. One block per token.
// ---------------------------------------------------------------------------
__global__ __launch_bounds__(64) void vq_finalize_kernel(
    const int* __restrict__ idx, const float* __restrict__ cb,
    float* __restrict__ out) {
  const int n = blockIdx.x;
  const int t = threadIdx.x;               // 64 threads, float4 over 256 elems
  const int k = idx[n];                    // block-uniform -> scalar load
  if (t == 0) out[(size_t)n * NCODE + k] = 1.0f;
  const float4* src = (const float4*)(cb + (size_t)k * EDIM);
  float4* dst = (float4*)(out + (size_t)NTOK * NCODE + (size_t)n * EDIM);
  dst[t] = src[t];
}

extern "C" void kernel_launch(void* const* d_in, const int* in_sizes, int n_in,
                              void* d_out, int out_size, void* d_ws, size_t ws_size,
                              hipStream_t stream) {
  const float* x  = (const float*)d_in[0];   // [16,1024,256]
  const float* cb = (const float*)d_in[1];   // [8192,256]
  const float* w  = (const float*)d_in[2];   // [256,128]
  const float* b  = (const float*)d_in[3];   // [128]
  float* out = (float*)d_out;                // discrete (134M) ++ quantized (4.2M)

  // Workspace layout (~12.7 MB)
  float* xp  = (float*)d_ws;                         // NTOK  * PDIM
  float* cbp = xp + (size_t)NTOK * PDIM;             // NCODE * PDIM
  float* cc  = cbp + (size_t)NCODE * PDIM;           // NCODE
  int*   idx = (int*)(cc + NCODE);                   // NTOK

  // Zero-fill can start first; it only touches d_out and overlaps nothing else
  // on this single stream, but keeps the launch queue busy front-to-back.
  const long long n4 = (long long)NTOK * NCODE / 4;  // 33,554,432 float4
  vq_fill_kernel<<<2048, 256, 0, stream>>>((float4*)out, n4);

  vq_project_kernel<<<NTOK + NCODE, PDIM, 0, stream>>>(x, cb, w, b, xp, cbp);
  vq_cc_kernel<<<NCODE, 32, 0, stream>>>(cbp, cc);
  vq_argmin_kernel<<<NTOK / 128, 256, 0, stream>>>(xp, cbp, cc, idx);
  vq_finalize_kernel<<<NTOK, 64, 0, stream>>>(idx, cb, out);
}